// TtLlamaRotary_32581621907457
// MI455X (gfx1250) — compile-verified
//
#include <hip/hip_runtime.h>

typedef __attribute__((ext_vector_type(2))) float v2f;
typedef __attribute__((ext_vector_type(8))) float v8f;

// Problem constants (from the reference): 8 q-heads, 1 k-head, S=128, D=128.
constexpr int NQ  = 8;
constexpr int S   = 128;
constexpr int HD  = 128;
constexpr int LDA = 132; // padded LDS row stride (dwords): banks (4m+k) mod 64,
                         // low-half/high-half lane windows disjoint -> no conflicts

__global__ __launch_bounds__(256) void rotary_wmma_f32_kernel(
    const float* __restrict__ xq,   // [1, 8, 128, 128]
    const float* __restrict__ xk,   // [1, 1, 128, 128]
    const float* __restrict__ rot,  // [1, 128, 128, 128]
    float* __restrict__ out)        // [8*128*128] q then [128*128] k
{
  __shared__ float lsA[16 * LDA];   // 16 x 128 A tile (rows 9..15 zero), 8448 B

  const int s    = blockIdx.x;      // sequence position
  const int tid  = threadIdx.x;
  const int wave = tid >> 5;        // 0..7 -> N tile
  const int lane = tid & 31;

  // ---- Stage A into LDS: rows 0..7 = xq heads, row 8 = xk, rows 9..15 = 0 ----
  for (int idx = tid; idx < 16 * HD; idx += 256) {
    const int row = idx >> 7;        // 0..15
    const int col = idx & (HD - 1);  // 0..127
    float v = 0.0f;
    if (row < NQ)       v = xq[(row * S + s) * HD + col];
    else if (row == NQ) v = xk[s * HD + col];
    lsA[row * LDA + col] = v;
  }
  __syncthreads();

  // ---- WMMA fragment indexing (16x16x4 f32, wave32) ----
  const int lh   = lane & 15;        // M (for A) / N column (for B,D)
  const int koff = (lane >> 4) << 1; // lanes 0-15: K=k0,k0+1 ; lanes 16-31: K=k0+2,k0+3
  const int nb   = wave << 4;        // N tile base
  const float* rotS = rot + (size_t)s * HD * HD;

  v8f c = {0.f, 0.f, 0.f, 0.f, 0.f, 0.f, 0.f, 0.f};

#pragma unroll
  for (int k0 = 0; k0 < HD; k0 += 4) {
    const int ka = k0 + koff;                       // even -> 8B aligned in LDS
    v2f a = *(const v2f*)(&lsA[lh * LDA + ka]);     // one ds_load_b64, bank-clean
    v2f b;
    b.x = rotS[(size_t)ka       * HD + nb + lh];    // rot[s][k][n]: 64B segments
    b.y = rotS[(size_t)(ka + 1) * HD + nb + lh];
    // D = A(16x4) * B(4x16) + C ; accumulate over K
    c = __builtin_amdgcn_wmma_f32_16x16x4_f32(
        /*neg_a=*/false, a, /*neg_b=*/false, b,
        /*c_mod=*/(short)0, c, /*reuse_a=*/false, /*reuse_b=*/false);
  }

  // ---- Store D. VGPR r: lanes 0-15 -> M=r (q heads), lanes 16-31 -> M=r+8 ----
  const int n = nb + lh;
  if (lane < 16) {
#pragma unroll
    for (int r = 0; r < 8; ++r)
      out[(r * S + s) * HD + n] = c[r];             // q head r, position s
  } else {
    out[NQ * S * HD + s * HD + n] = c[0];           // M=8 -> k head output
  }
  // M=9..15 (zero padding) intentionally discarded.
}

extern "C" void kernel_launch(void* const* d_in, const int* in_sizes, int n_in,
                              void* d_out, int out_size, void* d_ws, size_t ws_size,
                              hipStream_t stream) {
  (void)in_sizes; (void)n_in; (void)out_size; (void)d_ws; (void)ws_size;
  const float* xq  = (const float*)d_in[0];   // [1,8,128,128]
  const float* xk  = (const float*)d_in[1];   // [1,1,128,128]
  const float* rot = (const float*)d_in[2];   // [1,128,128,128]
  float* out = (float*)d_out;                 // 8*128*128 + 128*128 floats

  rotary_wmma_f32_kernel<<<S, 256, 0, stream>>>(xq, xk, rot, out);
}